// DevConvLayer_21260088115929
// MI455X (gfx1250) — compile-verified
//
#include <hip/hip_runtime.h>
#include <stdint.h>

// ---------------------------------------------------------------------------
// DevConvLayer reduction for MI455X (gfx1250, wave32).
//
// reference math:
//   s[j]    = sum_k x[j,k]
//   wmax[j] = max_c W_phi[c,j] ; wmin[j] = min_c W_phi[c,j]
//   t       = A[i,j] ? (s[i]-s[j]) : 0
//   M[i]    = max_j max(t*wmax[j], t*wmin[j])
//   dev[i]  = (row i has any neighbor) ? M[i] : 0 ; broadcast to 3 channels
//
// HBM-bound: 256 MB adjacency stream -> ~11 us floor at 23.3 TB/s.
// Data path: per-wave double-buffered GLOBAL_LOAD_ASYNC_TO_LDS_B128
// (ASYNCcnt) pipeline with TH_LOAD_NT (read-once stream must not evict the
// L2-resident s/wmax/wmin vectors), explicit s_wait_asynccnt / s_wait_dscnt
// fencing per the CDNA5 counter-ordering rules.
// ---------------------------------------------------------------------------

#define TPB        256
#define WAVE       32
#define WAVES      (TPB / WAVE)     // 8 waves per block
#define CHUNK_INTS 128              // per-wave chunk: 32 lanes * 4 int32 = 512 B

__global__ void dcl_precompute(const float* __restrict__ x,
                               const float* __restrict__ Wphi,
                               float* __restrict__ S,
                               float* __restrict__ WMX,
                               float* __restrict__ WMN,
                               int n) {
    int j = blockIdx.x * blockDim.x + threadIdx.x;
    if (j < n) {
        S[j] = x[3 * j + 0] + x[3 * j + 1] + x[3 * j + 2];
        float w0 = Wphi[j], w1 = Wphi[n + j], w2 = Wphi[2 * n + j];
        WMX[j] = fmaxf(w0, fmaxf(w1, w2));
        WMN[j] = fminf(w0, fminf(w1, w2));
    }
}

// --- CDNA5 async global->LDS copy (per-lane 16B), ASYNCcnt-tracked, NT -----
__device__ __forceinline__ void async_ld_b128_nt(uint32_t lds_off, const void* gptr) {
    asm volatile("global_load_async_to_lds_b128 %0, %1, off th:TH_LOAD_NT"
                 :
                 : "v"(lds_off), "v"(gptr)
                 : "memory");
}
__device__ __forceinline__ void wait_async_le1() {
    asm volatile("s_wait_asynccnt 1" ::: "memory");
}
__device__ __forceinline__ void wait_async_0() {
    asm volatile("s_wait_asynccnt 0" ::: "memory");
}
__device__ __forceinline__ void wait_ds_0() {
    asm volatile("s_wait_dscnt 0" ::: "memory");
}

__global__ void __launch_bounds__(TPB)
dcl_row_reduce(const int* __restrict__ adj,
               const float* __restrict__ S,
               const float* __restrict__ WMX,
               const float* __restrict__ WMN,
               float* __restrict__ out,
               int n) {
    __shared__ int   buf[2][WAVES][CHUNK_INTS];   // 8 KB double buffer
    __shared__ float red_m[WAVES];
    __shared__ int   red_any[WAVES];

    const int i   = blockIdx.x;
    const int tid = threadIdx.x;
    const int w   = tid >> 5;    // wave id   (0..7)
    const int l   = tid & 31;    // lane id   (0..31)

    const int per_wave = n / WAVES;               // 1024 cols per wave
    const int chunks   = per_wave / CHUNK_INTS;   // 8 chunks per wave
    const int base     = w * per_wave;

    const float      s_i  = S[i];
    const int* const rowp = adj + (size_t)i * (size_t)n;

    // LDS byte offsets of this lane's 16B slot in each buffer.
    // (generic-pointer low 32 bits == workgroup-relative LDS offset)
    const uint32_t lds0 = (uint32_t)(uintptr_t)&buf[0][w][l * 4];
    const uint32_t lds1 = (uint32_t)(uintptr_t)&buf[1][w][l * 4];

    float m   = -INFINITY;
    int   any = 0;

    // prime the pipeline: chunk 0 -> buffer 0
    async_ld_b128_nt(lds0, rowp + base + l * 4);

    for (int c = 0; c < chunks; ++c) {
        if (c + 1 < chunks) {
            // prefetch next chunk into the other buffer (last read of that
            // buffer was fenced by wait_ds_0 in the previous iteration)
            async_ld_b128_nt(((c + 1) & 1) ? lds1 : lds0,
                             rowp + base + (c + 1) * CHUNK_INTS + l * 4);
            wait_async_le1();   // in-order completion => chunk c has landed
        } else {
            wait_async_0();
        }

        const int* bp = &buf[c & 1][w][l * 4];
        int a0 = bp[0], a1 = bp[1], a2 = bp[2], a3 = bp[3];
        wait_ds_0();            // DS reads drained before buffer is reused

        const int    j0  = base + c * CHUNK_INTS + l * 4;
        const float4 sv  = *(const float4*)(S   + j0);
        const float4 wxv = *(const float4*)(WMX + j0);
        const float4 wnv = *(const float4*)(WMN + j0);

        float t;
        t = a0 ? (s_i - sv.x) : 0.0f; m = fmaxf(m, fmaxf(t * wxv.x, t * wnv.x));
        t = a1 ? (s_i - sv.y) : 0.0f; m = fmaxf(m, fmaxf(t * wxv.y, t * wnv.y));
        t = a2 ? (s_i - sv.z) : 0.0f; m = fmaxf(m, fmaxf(t * wxv.z, t * wnv.z));
        t = a3 ? (s_i - sv.w) : 0.0f; m = fmaxf(m, fmaxf(t * wxv.w, t * wnv.w));
        any |= a0 | a1 | a2 | a3;
    }

    // wave32 butterfly reduction (max, any)
    for (int off = 16; off > 0; off >>= 1) {
        m    = fmaxf(m, __shfl_xor(m, off, WAVE));
        any |= __shfl_xor(any, off, WAVE);
    }
    if (l == 0) { red_m[w] = m; red_any[w] = any; }
    __syncthreads();

    if (tid == 0) {
        float mm = red_m[0];
        int   aa = red_any[0];
        #pragma unroll
        for (int k = 1; k < WAVES; ++k) {
            mm = fmaxf(mm, red_m[k]);
            aa |= red_any[k];
        }
        const float dev = aa ? mm : 0.0f;
        out[3 * i + 0] = dev;
        out[3 * i + 1] = dev;
        out[3 * i + 2] = dev;
    }
}

extern "C" void kernel_launch(void* const* d_in, const int* in_sizes, int n_in,
                              void* d_out, int out_size, void* d_ws, size_t ws_size,
                              hipStream_t stream) {
    const float* x    = (const float*)d_in[0];   // (N,3) f32
    const int*   adj  = (const int*)d_in[1];     // (N,N) i32
    const float* Wphi = (const float*)d_in[2];   // (3,N) f32
    (void)d_in[3];                               // W_theta unused (faithful)
    (void)n_in; (void)ws_size;

    const int n = in_sizes[0] / 3;               // N = 8192

    float* S   = (float*)d_ws;                   // [n]
    float* WMX = S + n;                          // [n]
    float* WMN = WMX + n;                        // [n]

    dcl_precompute<<<(n + TPB - 1) / TPB, TPB, 0, stream>>>(x, Wphi, S, WMX, WMN, n);
    dcl_row_reduce<<<n, TPB, 0, stream>>>(adj, S, WMX, WMN, (float*)d_out, n);
    (void)out_size;
}